// SFTLayer_torch_new_11115375362095
// MI455X (gfx1250) — compile-verified
//
#include <hip/hip_runtime.h>
#include <hip/hip_bf16.h>
#include <stddef.h>
#include <stdint.h>

typedef __attribute__((ext_vector_type(2))) float v2f;
typedef __attribute__((ext_vector_type(8))) float v8f;

#define LRELU 0.01f
#define TH 16
#define TW 16
#define CCH 16                 // channel chunk for halo staging
#define HALO 18                // TH+2
#define IN_CH_STRIDE 324       // 18*18
// Interleaved K-pair layouts: element (k, col) lives at [(k>>1)*STRIDE + col*2 + (k&1)].
// Row strides chosen so stride % 64 == 32 -> lanes 0-15 and 16-31 (rows kp, kp+1)
// cover disjoint halves of the 64 LDS banks for b64 accesses.
#define MID_STRIDE 544         // 2*256 + 32 pad, 32 k-pair rows   (69,632 B)
#define WA_STRIDE 160          // 2*64  + 32 pad, 32 k-pair rows   (20,480 B)

// ---------------------------------------------------------------------------
// Kernel 1: hypernetwork. One block per batch.
//   hidden = leaky_relu(rep @ W0^T)   [b,64]
//   scale  = hidden @ W1^T            [b,576]  -> d_ws
// ---------------------------------------------------------------------------
__global__ __launch_bounds__(256) void sft_hyper_kernel(
    const float* __restrict__ rep,   // [8,256]
    const float* __restrict__ W0,    // [64,256]
    const float* __restrict__ W1,    // [576,64]
    float* __restrict__ scale_out)   // [8,576]
{
  __shared__ float s_rep[256];
  __shared__ float s_hidden[64];
  const int b = blockIdx.x;
  const int t = threadIdx.x;

  s_rep[t] = rep[b * 256 + t];
  __syncthreads();

  if (t < 64) {
    float acc = 0.f;
    const float* w0r = &W0[t * 256];
    #pragma unroll 8
    for (int k = 0; k < 256; ++k) acc += s_rep[k] * w0r[k];
    s_hidden[t] = acc > 0.f ? acc : LRELU * acc;
  }
  __syncthreads();

  for (int i = t; i < 576; i += 256) {
    float acc = 0.f;
    const float* w1r = &W1[i * 64];
    #pragma unroll
    for (int c = 0; c < 64; ++c) acc += s_hidden[c] * w1r[c];
    scale_out[b * 576 + i] = acc;
  }
}

// ---------------------------------------------------------------------------
// Kernel 2: fused depthwise-3x3 (+leaky) and 1x1 conv (WMMA f32) (+bias).
// Block = 256 threads (8 waves), one 16x16 pixel tile of one batch.
// ---------------------------------------------------------------------------
__global__ __launch_bounds__(256) void sft_main_kernel(
    const float* __restrict__ feat,    // [8,64,256,256]
    const float* __restrict__ scale,   // [8,576] from kernel 1
    const float* __restrict__ conv_w,  // [64,64] (o,c)
    const float* __restrict__ conv_b,  // [64]
    float* __restrict__ out)           // [8,64,256,256]
{
  __shared__ float s_in[CCH * IN_CH_STRIDE];    // halo chunk [c][18*18]   20,736 B
  __shared__ float s_mid[32 * MID_STRIDE];      // dw out, k-pair interleaved
  __shared__ float s_wA[32 * WA_STRIDE];        // conv_w,  k-pair interleaved
  __shared__ float s_coef[64 * 9];              // per-channel 3x3 kernels
  __shared__ float s_bias[64];

  const int t  = threadIdx.x;
  const int b  = blockIdx.y;
  const int h0 = (blockIdx.x >> 4) * TH;
  const int w0 = (blockIdx.x & 15) * TW;
  const bool interior = (h0 > 0) && (h0 < 240) && (w0 > 0) && (w0 < 240);

  // --- stage weights / coeffs / bias into LDS ---
  for (int i = t; i < 64 * 64; i += 256) {
    const int o = i >> 6, c = i & 63;           // A[m=o][k=c]
    s_wA[(c >> 1) * WA_STRIDE + o * 2 + (c & 1)] = conv_w[i];
  }
  for (int i = t; i < 576; i += 256) s_coef[i] = scale[b * 576 + i];
  if (t < 64) s_bias[t] = conv_b[t];

  const int ph = t >> 4, pw = t & 15;           // this thread's pixel in tile
  const float* featb = feat + (size_t)b * 64 * 65536;

  // --- depthwise conv, 16 channels at a time ---
  for (int cc = 0; cc < 64; cc += CCH) {
    __syncthreads();  // previous chunk's s_in readers done (also covers staging 1st iter)
    if (interior) {
      // CDNA5 async copy: memory -> LDS directly, tracked by ASYNCcnt.
      const float* gsrc = featb + (size_t)cc * 65536 + (size_t)(h0 - 1) * 256 + (w0 - 1);
      for (int i = t; i < CCH * IN_CH_STRIDE; i += 256) {
        const int cl  = i / IN_CH_STRIDE;
        const int rem = i - cl * IN_CH_STRIDE;
        const int y = rem / HALO, x = rem - y * HALO;
        const float* g = gsrc + (size_t)cl * 65536 + (size_t)y * 256 + x;
        const unsigned lds = (unsigned)(uintptr_t)&s_in[i];  // low 32 bits = LDS offset
        asm volatile("global_load_async_to_lds_b32 %0, %1, off"
                     :: "v"(lds), "v"(g) : "memory");
      }
    } else {
      for (int i = t; i < CCH * IN_CH_STRIDE; i += 256) {
        const int cl  = i / IN_CH_STRIDE;
        const int rem = i - cl * IN_CH_STRIDE;
        const int y = rem / HALO, x = rem - y * HALO;
        const int gh = h0 - 1 + y, gw = w0 - 1 + x;
        float v = 0.f;
        if ((unsigned)gh < 256u && (unsigned)gw < 256u)
          v = featb[(size_t)(cc + cl) * 65536 + (size_t)gh * 256 + gw];
        s_in[i] = v;
      }
    }
    if (cc + CCH < 64)   // speculative prefetch of next chunk -> global_prefetch_b8
      __builtin_prefetch(&featb[(size_t)(cc + CCH) * 65536 + (size_t)h0 * 256 + w0], 0, 0);
    asm volatile("s_wait_asynccnt 0x0" ::: "memory");
    __syncthreads();

    #pragma unroll
    for (int cl = 0; cl < CCH; ++cl) {
      const int c = cc + cl;
      const float* sp = &s_in[cl * IN_CH_STRIDE + ph * HALO + pw];
      const float* kc = &s_coef[c * 9];
      float acc = 0.f;
      acc += sp[0]  * kc[0]; acc += sp[1]  * kc[1]; acc += sp[2]  * kc[2];
      acc += sp[18] * kc[3]; acc += sp[19] * kc[4]; acc += sp[20] * kc[5];
      acc += sp[36] * kc[6]; acc += sp[37] * kc[7]; acc += sp[38] * kc[8];
      acc = acc > 0.f ? acc : LRELU * acc;
      // B matrix element (k=c, pixel=t), k-pair interleaved
      s_mid[(c >> 1) * MID_STRIDE + t * 2 + (c & 1)] = acc;
    }
  }
  __syncthreads();

  // --- 1x1 conv: D[16 out-ch][16 px], K=64 via 16 x V_WMMA_F32_16X16X4_F32 ---
  const int wave  = t >> 5;
  const int lane  = t & 31;
  const int lhalf = lane >> 4;     // fragment half: rows k0+{0,1} vs k0+{2,3}
  const int l16   = lane & 15;
  const int Mb    = (wave & 3) * 16;        // fixed out-channel tile per wave
  const int nb    = (wave >> 2) * 8;        // contiguous pixel-tile group
  float* outb = out + (size_t)b * 64 * 65536;

  // Preload this wave's entire A operand: one b64 per k-step.
  v2f a_frag[16];
  #pragma unroll
  for (int kp = 0; kp < 16; ++kp)           // k-pair row = 2*kp + lhalf
    a_frag[kp] = *(const v2f*)&s_wA[(2 * kp + lhalf) * WA_STRIDE + (Mb + l16) * 2];

  for (int ti = 0; ti < 8; ++ti) {
    const int pb = (nb + ti) * 16;
    const int bcol = (pb + l16) * 2;
    v8f acc = {};
    #pragma unroll
    for (int kp = 0; kp < 16; ++kp) {
      const v2f Bv = *(const v2f*)&s_mid[(2 * kp + lhalf) * MID_STRIDE + bcol];
      acc = __builtin_amdgcn_wmma_f32_16x16x4_f32(
          /*neg_a=*/false, a_frag[kp], /*neg_b=*/false, Bv,
          /*c_mod=*/(short)0, acc, /*reuse_a=*/false, /*reuse_b=*/false);
    }
    // D layout: VGPR r -> M = Mb + r + 8*lhalf, N = pb + l16
    const int p  = pb + l16;
    const size_t pixoff = (size_t)(h0 + (p >> 4)) * 256 + (w0 + (p & 15));
    #pragma unroll
    for (int r = 0; r < 8; ++r) {
      const int m = Mb + r + lhalf * 8;
      outb[(size_t)m * 65536 + pixoff] = acc[r] + s_bias[m];
    }
  }
}

extern "C" void kernel_launch(void* const* d_in, const int* in_sizes, int n_in,
                              void* d_out, int out_size, void* d_ws, size_t ws_size,
                              hipStream_t stream) {
  const float* feat   = (const float*)d_in[0];   // [8,64,256,256]
  const float* rep    = (const float*)d_in[1];   // [8,256]
  const float* W0     = (const float*)d_in[2];   // [64,256]
  const float* W1     = (const float*)d_in[3];   // [576,64]
  const float* conv_w = (const float*)d_in[4];   // [64,64]
  const float* conv_b = (const float*)d_in[5];   // [64]
  float* out   = (float*)d_out;
  float* scale = (float*)d_ws;                   // [8,576]

  sft_hyper_kernel<<<dim3(8), dim3(256), 0, stream>>>(rep, W0, W1, scale);
  sft_main_kernel<<<dim3(256, 8), dim3(256), 0, stream>>>(feat, scale, conv_w, conv_b, out);
}